// SimpleGraphConv_7292854469247
// MI455X (gfx1250) — compile-verified
//
#include <hip/hip_runtime.h>

typedef float v2f __attribute__((ext_vector_type(2)));
typedef float v8f __attribute__((ext_vector_type(8)));

#define DIM 128
#define LDS_PITCH 132  // 132 % 64 == 4 -> col stride of 4 banks, conflict-free b64 reads

// One wave computes a 16x128 output tile for BOTH GEMMs (shared A fragments):
//   h_self = x @ W_self + b_self   -> written to d_out (scatter adds on top later)
//   h_nei  = x @ W_nei             -> written to workspace
__global__ __launch_bounds__(256) void gemm_dual_wmma(
    const float* __restrict__ x,
    const float* __restrict__ Wself,
    const float* __restrict__ bself,
    const float* __restrict__ Wnei,
    float* __restrict__ hself_out,
    float* __restrict__ hnei_out,
    int nrows)
{
    __shared__ float ldsS[DIM * LDS_PITCH];
    __shared__ float ldsN[DIM * LDS_PITCH];

    const int tid = threadIdx.x;

    // Cooperative transposed stage: lds[n*PITCH + k] = W[k*128 + n]
    for (int i = tid; i < DIM * DIM; i += 256) {
        int k = i >> 7;
        int n = i & 127;
        ldsS[n * LDS_PITCH + k] = Wself[i];
        ldsN[n * LDS_PITCH + k] = Wnei[i];
    }
    __syncthreads();

    const int wave = tid >> 5;
    const int lane = tid & 31;
    const int rowbase = blockIdx.x * 128 + wave * 16;
    const int lrow = lane & 15;          // M (A) / N (B,C,D) position
    const int kb   = (lane >> 4) * 2;    // K pair selected by lane half

    // Preload A fragments: a[kk] = { x[row][4*kk+kb], x[row][4*kk+kb+1] }
    int arow = rowbase + lrow;
    if (arow >= nrows) arow = nrows - 1;          // clamp (stores are predicated)
    const float* aptr = x + (long)arow * DIM + kb;
    v2f a[32];
#pragma unroll
    for (int kk = 0; kk < 32; ++kk)
        a[kk] = *(const v2f*)(aptr + kk * 4);

    const int col = lrow;
    const int m0  = (lane >> 4) * 8;

    for (int ct = 0; ct < 8; ++ct) {
        const int n = ct * 16 + col;
        const float bias = bself[n];
        v8f accS = { bias, bias, bias, bias, bias, bias, bias, bias };
        v8f accN = {};

        const int bbase = n * LDS_PITCH + kb;
#pragma unroll
        for (int kk = 0; kk < 32; ++kk) {
            v2f bs = *(const v2f*)&ldsS[bbase + kk * 4];
            v2f bn = *(const v2f*)&ldsN[bbase + kk * 4];
            accS = __builtin_amdgcn_wmma_f32_16x16x4_f32(
                false, a[kk], false, bs, (short)0, accS, false, false);
            accN = __builtin_amdgcn_wmma_f32_16x16x4_f32(
                false, a[kk], false, bn, (short)0, accN, false, false);
        }

        // D layout: VGPR r -> M = r + 8*(lane>=16), N = lane&15
#pragma unroll
        for (int r = 0; r < 8; ++r) {
            int row = rowbase + m0 + r;
            if (row < nrows) {
                hself_out[(long)row * DIM + n] = accS[r];
                hnei_out [(long)row * DIM + n] = accN[r];
            }
        }
    }
}

// 32 lanes per edge; each lane owns a float4 chunk of the 128-dim message.
__global__ __launch_bounds__(256) void edge_scatter(
    const float* __restrict__ hnei,
    const long long* __restrict__ eidx,   // [2, E] int64: row 0 = src, row 1 = dst
    const float* __restrict__ ew,
    float* __restrict__ out,
    int E)
{
    long gtid = (long)blockIdx.x * 256 + threadIdx.x;
    long e = gtid >> 5;
    if (e >= E) return;
    int c = (int)(gtid & 31);

    long src = eidx[e];
    long dst = eidx[(long)E + e];
    float w = ew[e];

    const float4 v = *(const float4*)(hnei + src * DIM + c * 4);
    float* o = out + dst * DIM + c * 4;
    unsafeAtomicAdd(o + 0, v.x * w);
    unsafeAtomicAdd(o + 1, v.y * w);
    unsafeAtomicAdd(o + 2, v.z * w);
    unsafeAtomicAdd(o + 3, v.w * w);
}

extern "C" void kernel_launch(void* const* d_in, const int* in_sizes, int n_in,
                              void* d_out, int out_size, void* d_ws, size_t ws_size,
                              hipStream_t stream)
{
    const float*     x     = (const float*)d_in[0];
    const long long* eidx  = (const long long*)d_in[1];
    const float*     ew    = (const float*)d_in[2];
    const float*     Wself = (const float*)d_in[3];
    const float*     bself = (const float*)d_in[4];
    const float*     Wnei  = (const float*)d_in[5];

    const int N = in_sizes[0] / DIM;   // 100000 nodes
    const int E = in_sizes[2];         // 1600000 edges

    float* out  = (float*)d_out;
    float* hnei = (float*)d_ws;        // N*DIM*4 = 51.2 MB scratch

    // 1) Dual GEMM: d_out = x@W_self + b_self ; ws = x@W_nei
    int gblocks = (N + 127) / 128;
    hipLaunchKernelGGL(gemm_dual_wmma, dim3(gblocks), dim3(256), 0, stream,
                       x, Wself, bself, Wnei, out, hnei, N);

    // 2) Gather + scale + scatter-add into d_out
    long total = (long)E * 32;
    int eblocks = (int)((total + 255) / 256);
    hipLaunchKernelGGL(edge_scatter, dim3(eblocks), dim3(256), 0, stream,
                       hnei, eidx, ew, out, E);
}